// adder2d_11785390260463
// MI455X (gfx1250) — compile-verified
//
#include <hip/hip_runtime.h>
#include <cstdint>

// AdderNet 2D (negative-L1 "conv"), MI455X / gfx1250.
//
// Roofline: ~26 MB traffic (~1.1us @ 23.3 TB/s) vs 3.7G f32 VALU ops ->
// strongly VALU-bound. Sum|a-b| is not bilinear, so WMMA cannot express it;
// the optimal CDNA5 recipe is: TDM DMA for the weight block, async-to-LDS
// copies for the input halo tile, and a fully unrolled f32 sub + add-|src|
// inner loop fed by ds_load_b128 broadcasts out of the 320KB WGP LDS.
//
// Tiling: 28x8 output tiles (56 = 2*28 = 7*8) -> exact coverage, zero wasted
// taps, 224 threads = 224 pixels. Each WG handles one co-half (32 channels),
// whose weights are a contiguous 73,728B block of `adder` -> perfect 1-D TDM.

namespace {
constexpr int kCin  = 64;
constexpr int kHW   = 56;
constexpr int kCout = 64;

constexpr int kTileW = 28, kTileH = 8;          // 224 pixels = 224 threads
constexpr int kHaloW = kTileW + 2;              // 30
constexpr int kHaloH = kTileH + 2;              // 10
constexpr int kF     = kCin * 9;                // 576 taps
constexpr int kCoH   = kCout / 2;               // 32 channels per WG

constexpr int kWrawF = kCoH * kF;               // 18432 floats (73,728 B)
constexpr int kWsTF  = kF * kCoH;               // 18432 floats (73,728 B)
constexpr int kXsF   = kCin * kHaloH * kHaloW;  // 19200 floats (76,800 B)
constexpr int kLdsBytes = (kWrawF + kWsTF + kXsF) * (int)sizeof(float); // 224,256
static_assert(kLdsBytes <= 320 * 1024, "fits CDNA5 WGP LDS");

typedef unsigned __attribute__((ext_vector_type(4))) v4u;
typedef unsigned __attribute__((ext_vector_type(8))) v8u;
}  // namespace

__global__ __launch_bounds__(224)
void adder2d_l1conv_kernel(const float* __restrict__ x,
                           const float* __restrict__ adder,
                           float* __restrict__ out) {
  extern __shared__ float smem[];
  float* wsraw = smem;                    // [32][576]  raw co-half weights (TDM dest)
  float* wsT   = smem + kWrawF;           // [576][32]  transposed for b128 broadcast
  float* xs    = smem + kWrawF + kWsTF;   // [64][10][30] input halo tile

  const int t    = threadIdx.x;           // 0..223
  const int half = blockIdx.z & 1;
  const int n    = blockIdx.z >> 1;
  const int co0  = half * kCoH;
  const int h0   = blockIdx.y * kTileH;
  const int w0   = blockIdx.x * kTileW;

  // ---- TDM: DMA contiguous 73,728B weight block adder[co0*576 ..) -> LDS ----
  // 1-D copy expressed as a 2D descriptor: tile_dim0 = tensor_dim0 =
  // tensor_dim0_stride = 18432 elements, data_size = 4B, tile_dim1 = 1.
  // TDM ignores EXEC, so issue from wave 0 only (uniform branch).
  if (t < 32) {
    const uint64_t ga  = (uint64_t)(uintptr_t)(adder + (size_t)co0 * kF);
    const unsigned lds = (unsigned)(uintptr_t)wsraw;   // LDS byte offset
    v4u g0;
    g0.x = 1u;                                          // count=1 (valid D#)
    g0.y = lds;                                         // lds_addr
    g0.z = (unsigned)(ga & 0xffffffffu);                // global_addr[31:0]
    g0.w = (unsigned)((ga >> 32) & 0x01ffffffu)         // global_addr[56:32]
           | (2u << 30);                                // type=2 ("image")
    v8u g1;
    g1.s0 = 2u << 16;           // workgroup_mask=0, data_size=2 (4 bytes)
    g1.s1 = 18432u << 16;       // tensor_dim0[15:0] @ bits63:48 (=0x4800)
    g1.s2 = 1u << 16;           // tensor_dim0 hi=0, tensor_dim1=1
    g1.s3 = 18432u << 16;       // tensor_dim1 hi=0, tile_dim0=18432
    g1.s4 = 1u;                 // tile_dim1=1, tile_dim2=0
    g1.s5 = 18432u;             // tensor_dim0_stride lo
    g1.s6 = 0u;                 // stride0 hi = 0, tensor_dim1_stride lo = 0
    g1.s7 = 0u;                 // tensor_dim1_stride hi = 0
    asm volatile("tensor_load_to_lds %0, %1" :: "s"(g0), "s"(g1) : "memory");
  }

  // ---- zero-fill halo tile (implements pad=1) ----
  for (int e = t; e < kXsF; e += 224) xs[e] = 0.0f;
  __syncthreads();  // zeros visible before async writes to same region

  // ---- CDNA5 async copy: interior of x tile -> LDS ----
  // GVS mode: SGPR 64-bit base (uniform in n) + per-lane 32-bit byte offset.
  const float* xbase = x + (size_t)n * kCin * kHW * kHW;
  for (int e = t; e < kXsF; e += 224) {
    const int ci = e / (kHaloH * kHaloW);
    const int rc = e - ci * (kHaloH * kHaloW);
    const int r  = rc / kHaloW;
    const int c  = rc - r * kHaloW;
    const int h  = h0 - 1 + r;
    const int w  = w0 - 1 + c;
    if ((unsigned)h < (unsigned)kHW && (unsigned)w < (unsigned)kHW) {
      const unsigned ldsb = (unsigned)(uintptr_t)(&xs[e]);
      const unsigned voff = (unsigned)(((ci * kHW + h) * kHW + w) * 4);
      asm volatile("global_load_async_to_lds_b32 %0, %1, %2"
                   :: "v"(ldsb), "v"(voff), "s"(xbase)
                   : "memory");
    }
  }
  asm volatile("s_wait_asynccnt 0" ::: "memory");  // this wave's async writes done
  __builtin_amdgcn_s_wait_tensorcnt(0);            // wave0's TDM done (0 elsewhere)
  __syncthreads();                                 // all LDS producers visible

  // ---- LDS->LDS transpose: wsT[f*32 + j] = wsraw[j*576 + f] ----
  // (reads conflict-free/contiguous; one-time cost, negligible vs compute)
  for (int e = t; e < kWrawF; e += 224) {
    const int j = e / kF;
    const int f = e - j * kF;
    wsT[f * kCoH + j] = wsraw[e];
  }
  __syncthreads();

  // ---- compute: one pixel per thread, 32 output channels ----
  const int py = t / kTileW;     // 0..7
  const int px = t - py * kTileW;  // 0..27

  float acc[kCoH];
#pragma unroll
  for (int j = 0; j < kCoH; ++j) acc[j] = 0.0f;

  for (int ci = 0; ci < kCin; ++ci) {
    const float* xrow = &xs[ci * (kHaloH * kHaloW)];
#pragma unroll
    for (int kh = 0; kh < 3; ++kh) {
      // 3 x-values reused across kw
      const float xv0 = xrow[(py + kh) * kHaloW + px + 0];
      const float xv1 = xrow[(py + kh) * kHaloW + px + 1];
      const float xv2 = xrow[(py + kh) * kHaloW + px + 2];
#pragma unroll
      for (int kw = 0; kw < 3; ++kw) {
        const float a = (kw == 0) ? xv0 : (kw == 1) ? xv1 : xv2;
        const int f = (ci * 3 + kh) * 3 + kw;
        // 32 contiguous weights per tap: 8x ds_load_b128, uniform address
        // across the wave -> LDS broadcast, conflict-free.
        const float4* wq = reinterpret_cast<const float4*>(&wsT[f * kCoH]);
#pragma unroll
        for (int q = 0; q < kCoH / 4; ++q) {
          const float4 wv = wq[q];
          // v_sub_f32 + v_add_f32 |src|: 2 VALU ops per tap, 32 independent
          // accumulation chains -> ample ILP for dual-issue.
          acc[4 * q + 0] += __builtin_fabsf(a - wv.x);
          acc[4 * q + 1] += __builtin_fabsf(a - wv.y);
          acc[4 * q + 2] += __builtin_fabsf(a - wv.z);
          acc[4 * q + 3] += __builtin_fabsf(a - wv.w);
        }
      }
    }
  }

  // ---- store: exact tiling -> no guards; lanes = consecutive w -> coalesced ----
  const int h = h0 + py;
  const int w = w0 + px;
  float* op = out + (((size_t)n * kCout + co0) * (kHW * kHW)) + h * kHW + w;
#pragma unroll
  for (int j = 0; j < kCoH; ++j) {
    op[(size_t)j * (kHW * kHW)] = -acc[j];
  }
}

extern "C" void kernel_launch(void* const* d_in, const int* in_sizes, int n_in,
                              void* d_out, int out_size, void* d_ws, size_t ws_size,
                              hipStream_t stream) {
  (void)in_sizes; (void)n_in; (void)out_size; (void)d_ws; (void)ws_size;
  const float* x     = (const float*)d_in[0];
  const float* adder = (const float*)d_in[1];
  float* out         = (float*)d_out;

  // 224,256B dynamic LDS (> default cap; CDNA5 WGP has 320KB)
  hipFuncSetAttribute(reinterpret_cast<const void*>(adder2d_l1conv_kernel),
                      hipFuncAttributeMaxDynamicSharedMemorySize, kLdsBytes);

  dim3 grid(kHW / kTileW,        // 2
            kHW / kTileH,        // 7
            16 * 2);             // n * co-half -> 448 workgroups
  adder2d_l1conv_kernel<<<grid, dim3(224, 1, 1), kLdsBytes, stream>>>(x, adder, out);
}